// AnalyticLens_68289980006590
// MI455X (gfx1250) — compile-verified
//
#include <hip/hip_runtime.h>

typedef float v2f __attribute__((ext_vector_type(2)));
typedef float v8f __attribute__((ext_vector_type(8)));

namespace {
constexpr int   NPIX_LO   = 128;
constexpr int   NV_HI_C   = 256;     // hi-res velocity bins
constexpr int   PWG       = 128;     // lo-res pixels (= threads) per workgroup
constexpr float PS_HI     = 0.0125f;     // PIXSCALE_HI = 0.05/4
constexpr float FOV_HALF  = 3.19375f;    // 0.5*(512-1)*PS_HI
constexpr float VEL0_HI_C = -323.75f;    // -320 - 0.5*(10 - 2.5)
constexpr float INV_DV_HI = 0.4f;        // 1/2.5
}

// One thread = one lo-res pixel. Private 256-bin velocity histogram per thread
// in LDS (layout hist[vbin*PWG + tid] -> conflict-free for wave32).
// Final stage: out(64 x P) = M(64 x 256) @ hist(256 x P) on the WMMA pipe,
// where M[vlo,vhi] = 1/64 iff vhi>>2 == vlo (4:1 velocity mean fused with the
// 1/16 spatial mean already accumulated into the histogram).
__global__ __launch_bounds__(PWG)
void AnalyticLens_68289980006590_kernel(
    const float* __restrict__ p_inc,  const float* __restrict__ p_rot,
    const float* __restrict__ p_lb,   const float* __restrict__ p_vsh,
    const float* __restrict__ p_x0,   const float* __restrict__ p_y0,
    const float* __restrict__ p_dist, const float* __restrict__ p_thE,
    const float* __restrict__ p_I0,   const float* __restrict__ p_Rd,
    const float* __restrict__ p_vmax, const float* __restrict__ p_Rt,
    float* __restrict__ out)
{
  extern __shared__ float hist[];               // [NV_HI_C][PWG] floats
  const int tid = threadIdx.x;
  const int wg  = blockIdx.x;

  // ---- zero this thread's histogram column (no cross-thread sharing yet) ----
  #pragma unroll 8
  for (int v = 0; v < NV_HI_C; ++v) hist[v * PWG + tid] = 0.0f;

  // ---- scalar parameters (uniform loads) ----
  const float inc   = *p_inc;
  const float rot   = *p_rot;
  const float lb    = *p_lb;
  const float vsh   = *p_vsh;
  const float x0    = *p_x0;
  const float y0    = *p_y0;
  const float dist  = *p_dist;
  const float thE   = *p_thE;
  const float I0    = *p_I0;
  const float Rd    = *p_Rd;
  const float vmax  = *p_vmax;
  const float Rt    = *p_Rt;

  const float cos_i   = __cosf(inc);
  const float sin_i   = __sinf(inc);
  const float pa      = rot + 1.57079632679489662f;
  const float cp      = __cosf(pa);
  const float sp      = __sinf(pa);
  const float inv_ci  = 1.0f / (cos_i + 1e-12f);
  const float sigma   = fabsf(lb) + 1e-12f;
  const float pc_scl  = dist * (1.0f / 206265.0f);   // 1/arcsec_per_pc
  const float inv_Rd  = 1.0f / Rd;
  const float inv_Rt  = 1.0f / Rt;
  const float vfac    = vmax * 0.63661977236758134f; // vmax * 2/pi

  // sqrt(2)*erfinv(2*p-1) for p=(k+0.5)/8 : compile-time normal quantiles
  const float unit[8] = {
    -1.53412054f, -0.88714656f, -0.48877641f, -0.15731068f,
     0.15731068f,  0.48877641f,  0.88714656f,  1.53412054f };

  // ---- this thread's lo-res pixel ----
  const int p   = wg * PWG + tid;
  const int xlo = p & (NPIX_LO - 1);
  const int ylo = p >> 7;

  // ---- 4x4 hi-res sub-pixels: raytrace + models + 8 quantile deposits ----
  for (int dy = 0; dy < 4; ++dy) {
    const float thy = -FOV_HALF + PS_HI * (float)(ylo * 4 + dy);
    for (int dx = 0; dx < 4; ++dx) {
      const float thx = -FOV_HALF + PS_HI * (float)(xlo * 4 + dx);
      // SIS deflection
      const float r    = sqrtf(thx * thx + thy * thy) + 1e-12f;
      const float defl = thE / r;
      const float bx   = thx - defl * thx;
      const float by   = thy - defl * thy;
      // sky -> intrinsic disk plane
      const float X  = (bx - x0) * pc_scl;
      const float Y  = (by - y0) * pc_scl;
      const float xg =  cp * X + sp * Y;
      const float yg = (-sp * X + cp * Y) * inv_ci;
      const float R  = sqrtf(xg * xg + yg * yg);
      // intensity (pre-divided by K=8) and LOS velocity
      const float Iw   = I0 * __expf(-R * inv_Rd) * 0.125f;
      const float vlos = vfac * atanf(R * inv_Rt) * sin_i * (xg / (R + 1e-12f)) + vsh;
      #pragma unroll
      for (int k = 0; k < 8; ++k) {
        const float vs  = vlos + sigma * unit[k];
        const float ivf = (vs - VEL0_HI_C) * INV_DV_HI;
        int iv0 = (int)floorf(ivf);
        iv0 = iv0 < 0 ? 0 : (iv0 > NV_HI_C - 1 ? NV_HI_C - 1 : iv0);
        const int iv1 = (iv0 + 1 > NV_HI_C - 1) ? NV_HI_C - 1 : iv0 + 1;
        float fv = ivf - (float)iv0;
        fv = fv < 0.0f ? 0.0f : (fv > 1.0f ? 1.0f : fv);
        hist[iv0 * PWG + tid] += (1.0f - fv) * Iw;   // own column: race-free
        hist[iv1 * PWG + tid] += fv * Iw;
      }
    }
  }

  __syncthreads();

  // ---- WMMA stage: out(64 x 128) = M(64 x 256) @ hist(256 x 128), /64 folded into M
  // wave w handles lo-bin group [16w, 16w+16); loops over 8 pixel tiles of 16.
  const int lane  = tid & 31;
  const int wave  = tid >> 5;          // 0..3
  const int mrow  = lane & 15;         // A: M row / B,D: N column
  const int khalf = lane >> 4;         // 0 -> K{0,1}, 1 -> K{2,3}
  const int vbase = wave * 64;         // first hi bin feeding this vlo group

  for (int pt = 0; pt < 8; ++pt) {
    const int pixN = pt * 16 + mrow;   // local pixel column for B / D
    v8f c = {0.f, 0.f, 0.f, 0.f, 0.f, 0.f, 0.f, 0.f};
    #pragma unroll
    for (int kc = 0; kc < 16; ++kc) {
      // A tile (16x4 of M): only row m==kc nonzero, value 1/64 (both K slots)
      const float aval = (mrow == kc) ? (1.0f / 64.0f) : 0.0f;
      v2f a; a.x = aval; a.y = aval;
      // B tile (4x16 of hist): rows K = 4*kc + khalf*2 + {0,1}
      const int krow = vbase + 4 * kc + khalf * 2;
      v2f b;
      b.x = hist[(krow    ) * PWG + pixN];
      b.y = hist[(krow + 1) * PWG + pixN];
      c = __builtin_amdgcn_wmma_f32_16x16x4_f32(
              false, a, false, b, (short)0, c, false, false);
    }
    // D layout: VGPR i holds row M = i + 8*khalf, column N = mrow
    const int pix_g = wg * PWG + pt * 16 + mrow;
    #pragma unroll
    for (int i = 0; i < 8; ++i) {
      const int vlo = wave * 16 + i + 8 * khalf;
      out[vlo * (NPIX_LO * NPIX_LO) + pix_g] = c[i];
    }
  }
}

extern "C" void kernel_launch(void* const* d_in, const int* in_sizes, int n_in,
                              void* d_out, int out_size, void* d_ws, size_t ws_size,
                              hipStream_t stream) {
  (void)in_sizes; (void)n_in; (void)out_size; (void)d_ws; (void)ws_size;
  const float* inc  = (const float*)d_in[0];
  const float* rot  = (const float*)d_in[1];
  const float* lb   = (const float*)d_in[2];
  const float* vsh  = (const float*)d_in[3];
  const float* x0   = (const float*)d_in[4];
  const float* y0   = (const float*)d_in[5];
  const float* dist = (const float*)d_in[6];
  const float* thE  = (const float*)d_in[7];
  const float* I0   = (const float*)d_in[8];
  const float* Rd   = (const float*)d_in[9];
  const float* vmax = (const float*)d_in[10];
  const float* Rt   = (const float*)d_in[11];
  float* out = (float*)d_out;

  const int    nblocks = (NPIX_LO * NPIX_LO) / PWG;          // 128
  const size_t smem    = (size_t)NV_HI_C * PWG * sizeof(float); // 128 KB
  AnalyticLens_68289980006590_kernel<<<nblocks, PWG, smem, stream>>>(
      inc, rot, lb, vsh, x0, y0, dist, thE, I0, Rd, vmax, Rt, out);
}